// SDGCN32_3496103379559
// MI455X (gfx1250) — compile-verified
//
#include <hip/hip_runtime.h>
#include <hip/hip_bf16.h>

// ---------------------------------------------------------------------------
// SDGCN32 for MI455X (gfx1250, wave32).
// GEMMs (x@W1 [K=64], x@Wc [K=32]) run on V_WMMA_F32_16X16X4_F32: one wave
// computes a 16x32 f32 tile with two v8f accumulators. Everything else is
// the L2-resident edge scatter (the real bottleneck) + wave32 shuffle
// reductions for the per-row min-max norms and the folded feats@W2.
// ---------------------------------------------------------------------------

typedef __attribute__((ext_vector_type(2))) float v2f;
typedef __attribute__((ext_vector_type(8))) float v8f;

#define WAVE 32

// ---------------------------------------------------------------------------
// H[n,32] = A[n,K] @ W[K,32]   (row-major, K in {32,64}).
// One wave per 16-row tile; EXEC stays all-ones through the WMMA region
// (wave-uniform exit; out-of-range rows are CLAMPED, not predicated, since
// their results are never stored -- avoids exec save/restore per load and
// lets the A pair load become a single global_load_b64).
// ---------------------------------------------------------------------------
__global__ void gemm_nodes_wmma(const float* __restrict__ A,
                                const float* __restrict__ W,
                                float* __restrict__ H,
                                int n_nodes, int K) {
    int gtid = blockIdx.x * blockDim.x + threadIdx.x;
    int wave = gtid >> 5;           // wave-uniform
    int lane = threadIdx.x & 31;
    int row0 = wave * 16;
    if (row0 >= n_nodes) return;    // wave-uniform exit

    int m    = lane & 15;           // row within 16-row tile
    int half = lane >> 4;           // 0: K pair {0,1}; 1: K pair {2,3}
    int koff = half * 2;
    int nc   = lane & 15;           // column within 16-col tile

    int r = row0 + m;
    if (r >= n_nodes) r = n_nodes - 1;      // clamp: safe, store is guarded
    const float* arow = A + (size_t)r * K;

    v8f c0 = {};                    // columns 0..15
    v8f c1 = {};                    // columns 16..31

#pragma unroll 4
    for (int kb = 0; kb < K; kb += 4) {
        v2f a = *(const v2f*)(arow + kb + koff);   // 8B-aligned pair load
        v2f b0, b1;
        b0.x = W[(kb + koff)     * 32 + nc];
        b0.y = W[(kb + koff + 1) * 32 + nc];
        b1.x = W[(kb + koff)     * 32 + 16 + nc];
        b1.y = W[(kb + koff + 1) * 32 + 16 + nc];
#if __has_builtin(__builtin_amdgcn_wmma_f32_16x16x4_f32)
        c0 = __builtin_amdgcn_wmma_f32_16x16x4_f32(false, a, false, b0,
                                                   (short)0, c0, false, false);
        c1 = __builtin_amdgcn_wmma_f32_16x16x4_f32(false, a, false, b1,
                                                   (short)0, c1, false, false);
#else
        for (int rr = 0; rr < 8; ++rr) {
            int mr = row0 + half * 8 + rr;
            const float* ar = A + (size_t)(mr < n_nodes ? mr : 0) * K;
            float s0 = c0[rr], s1 = c1[rr];
            for (int kk = 0; kk < 4; ++kk) {
                float av = ar[kb + kk];
                s0 += av * W[(kb + kk) * 32 + nc];
                s1 += av * W[(kb + kk) * 32 + 16 + nc];
            }
            c0[rr] = s0; c1[rr] = s1;
        }
#endif
    }

    // C/D layout: VGPR rr, lanes 0-15 -> M=rr, lanes 16-31 -> M=rr+8.
    int mbase = row0 + half * 8;
    for (int rr = 0; rr < 8; ++rr) {
        int rrow = mbase + rr;
        if (rrow < n_nodes) {
            H[(size_t)rrow * 32 + nc]      = c0[rr];
            H[(size_t)rrow * 32 + 16 + nc] = c1[rr];
        }
    }
}

// ---------------------------------------------------------------------------
__global__ void fill0(float* __restrict__ p, int n) {
    int i = blockIdx.x * blockDim.x + threadIdx.x;
    if (i < n) p[i] = 0.0f;
}

__global__ void deg_count(const int* __restrict__ dst, float* __restrict__ deg, int E) {
    int e = blockIdx.x * blockDim.x + threadIdx.x;
    if (e < E) atomicAdd(&deg[dst[e]], 1.0f);
}

__global__ void deg_to_dis(const float* __restrict__ deg, float* __restrict__ dis,
                           float* __restrict__ snorm, int n) {
    int i = blockIdx.x * blockDim.x + threadIdx.x;
    if (i < n) {
        float s = rsqrtf(deg[i] + 1.0f);
        dis[i]   = s;
        snorm[i] = s * s;
    }
}

__global__ void edge_norm(const int* __restrict__ src, const int* __restrict__ dst,
                          const float* __restrict__ dis, float* __restrict__ enorm, int E) {
    int e = blockIdx.x * blockDim.x + threadIdx.x;
    if (e < E) enorm[e] = dis[src[e]] * dis[dst[e]];
}

// agg[dst] += h[src] * enorm ; one wave per edge: 32 lanes cover the 32
// channels of one dst row (coalesced gather + one 128B atomic burst).
__global__ void scatter_edges(const float* __restrict__ h,
                              const int* __restrict__ src,
                              const int* __restrict__ dst,
                              const float* __restrict__ enorm,
                              float* __restrict__ agg,
                              unsigned int total) {
    unsigned int gid = blockIdx.x * blockDim.x + threadIdx.x;
    if (gid >= total) return;
    int e = gid >> 5;
    int c = gid & 31;
    float v = h[(size_t)src[e] * 32 + c] * enorm[e];
    atomicAdd(&agg[(size_t)dst[e] * 32 + c], v);
}

// agg = agg + snorm * h + bc
__global__ void gcn_finalize(float* __restrict__ agg, const float* __restrict__ h,
                             const float* __restrict__ snorm,
                             const float* __restrict__ bc, int n_nodes) {
    int gid = blockIdx.x * blockDim.x + threadIdx.x;
    if (gid >= n_nodes * 32) return;
    int node = gid >> 5;
    int c    = gid & 31;
    agg[gid] = agg[gid] + snorm[node] * h[gid] + bc[c];
}

__device__ __forceinline__ void wave_minmax(float t, float& mn, float& mx) {
    mn = t; mx = t;
    for (int i = 16; i >= 1; i >>= 1) {
        mn = fminf(mn, __shfl_xor(mn, i, WAVE));
        mx = fmaxf(mx, __shfl_xor(mx, i, WAVE));
    }
}

// x0 = mynorm(relu(h1 + b1)); nbuf0 = mynorm(x0); out = b2 + x0 @ W2[0:32]
__global__ void init_feat(const float* __restrict__ h1, const float* __restrict__ b1,
                          const float* __restrict__ W2, const float* __restrict__ b2,
                          float* __restrict__ x0, float* __restrict__ nbuf0,
                          float* __restrict__ out, int n_nodes) {
    int gid  = blockIdx.x * blockDim.x + threadIdx.x;
    int node = gid >> 5;
    int c    = gid & 31;
    if (node >= n_nodes) return;

    float t = h1[(size_t)node * 32 + c] + b1[c];
    t = fmaxf(t, 0.0f);                       // relu
    float mn, mx;
    wave_minmax(t, mn, mx);
    float v0 = 2.0f * (t - mn) / (mx - mn + 1e-8f) - 1.0f;   // x0
    x0[(size_t)node * 32 + c] = v0;

    wave_minmax(v0, mn, mx);
    float n0 = 2.0f * (v0 - mn) / (mx - mn + 1e-8f) - 1.0f;  // norms[0]
    nbuf0[(size_t)node * 32 + c] = n0;

    // out = b2 + x0 @ W2[0:32,:]
    for (int j = 0; j < 10; ++j) {
        float p = v0 * W2[c * 10 + j];
        for (int i = 16; i >= 1; i >>= 1) p += __shfl_xor(p, i, WAVE);
        if (c == 0) out[(size_t)node * 10 + j] = b2[j] + p;
    }
}

// part = raw ? cur : (mynorm(cur) - nbuf_old); nbuf = mynorm(cur);
// out += part @ W2part
__global__ void feat_accum(const float* __restrict__ cur, float* __restrict__ nbuf,
                           const float* __restrict__ W2part, float* __restrict__ out,
                           int n_nodes, int raw_part) {
    int gid  = blockIdx.x * blockDim.x + threadIdx.x;
    int node = gid >> 5;
    int c    = gid & 31;
    if (node >= n_nodes) return;

    float t = cur[(size_t)node * 32 + c];
    float mn, mx;
    wave_minmax(t, mn, mx);
    float v = 2.0f * (t - mn) / (mx - mn + 1e-8f) - 1.0f;
    float part = raw_part ? t : (v - nbuf[(size_t)node * 32 + c]);
    nbuf[(size_t)node * 32 + c] = v;

    for (int j = 0; j < 10; ++j) {
        float p = part * W2part[c * 10 + j];
        for (int i = 16; i >= 1; i >>= 1) p += __shfl_xor(p, i, WAVE);
        if (c == 0) out[(size_t)node * 10 + j] += p;
    }
}

// ---------------------------------------------------------------------------
extern "C" void kernel_launch(void* const* d_in, const int* in_sizes, int n_in,
                              void* d_out, int out_size, void* d_ws, size_t ws_size,
                              hipStream_t stream) {
    const float* x    = (const float*)d_in[0];
    const int*   edge = (const int*)d_in[1];     // jnp int64 -> int32 (x64 disabled)
    const float* W1   = (const float*)d_in[2];
    const float* b1   = (const float*)d_in[3];
    const float* Wc   = (const float*)d_in[4];
    const float* bc   = (const float*)d_in[5];
    const float* W2   = (const float*)d_in[6];
    const float* b2   = (const float*)d_in[7];

    const int N = in_sizes[0] / 64;
    const int E = in_sizes[1] / 2;
    const int L = in_sizes[4] / (32 * 32);
    const int* srcI = edge;
    const int* dstI = edge + E;
    float* out = (float*)d_out;

    // workspace carve-up (floats)
    float* ws    = (float*)d_ws;
    float* deg   = ws;  ws += N;
    float* dis   = ws;  ws += N;
    float* snorm = ws;  ws += N;
    float* enorm = ws;  ws += E;
    float* bufA  = ws;  ws += (size_t)N * 32;
    float* bufB  = ws;  ws += (size_t)N * 32;
    float* hbuf  = ws;  ws += (size_t)N * 32;
    float* nb0   = ws;  ws += (size_t)N * 32;
    float* nb1   = ws;  ws += (size_t)N * 32;
    float* nbuf[2] = { nb0, nb1 };

    const int BLK = 256;
    auto cdiv = [](long long a, long long b) { return (int)((a + b - 1) / b); };

    // ---- normalization coefficients ----
    fill0<<<cdiv(N, BLK), BLK, 0, stream>>>(deg, N);
    deg_count<<<cdiv(E, BLK), BLK, 0, stream>>>(dstI, deg, E);
    deg_to_dis<<<cdiv(N, BLK), BLK, 0, stream>>>(deg, dis, snorm, N);
    edge_norm<<<cdiv(E, BLK), BLK, 0, stream>>>(srcI, dstI, dis, enorm, E);

    // ---- x0 = mynorm(relu(x @ W1 + b1)); out = b2 + part0 ----
    {
        int waves = (N + 15) / 16;
        gemm_nodes_wmma<<<cdiv((long long)waves * 32, BLK), BLK, 0, stream>>>(
            x, W1, hbuf, N, 64);
        init_feat<<<cdiv((long long)N * 32, BLK), BLK, 0, stream>>>(
            hbuf, b1, W2, b2, bufA, nb0, out, N);
    }

    // ---- 31 GCN layers with folded feature/W2 accumulation ----
    float* cur   = bufA;
    float* other = bufB;
    const unsigned int scat_total = (unsigned int)E * 32u;
    for (int i = 1; i <= L; ++i) {
        int waves = (N + 15) / 16;
        gemm_nodes_wmma<<<cdiv((long long)waves * 32, BLK), BLK, 0, stream>>>(
            cur, Wc + (size_t)(i - 1) * 32 * 32, hbuf, N, 32);
        fill0<<<cdiv((long long)N * 32, BLK), BLK, 0, stream>>>(other, N * 32);
        scatter_edges<<<cdiv(scat_total, BLK), BLK, 0, stream>>>(
            hbuf, srcI, dstI, enorm, other, scat_total);
        gcn_finalize<<<cdiv((long long)N * 32, BLK), BLK, 0, stream>>>(
            other, hbuf, snorm, bc + (size_t)(i - 1) * 32, N);
        int raw = (i == 1 || i == 16) ? 1 : 0;
        feat_accum<<<cdiv((long long)N * 32, BLK), BLK, 0, stream>>>(
            other, nbuf[i & 1], W2 + (size_t)32 * i * 10, out, N, raw);
        float* tmp = cur; cur = other; other = tmp;
    }
}